// RegionTarget_55181739819592
// MI455X (gfx1250) — compile-verified
//
#include <hip/hip_runtime.h>
#include <stdint.h>

// Problem constants (match reference setup_inputs)
#define BB 32
#define AA 5
#define HH 104
#define WW 104
#define TT 30
#define HW (HH * WW)          // 10816
#define N2 (BB * 2 * AA * HW) // 3,461,120  (B,2A,H,W) planes
#define N1 (BB * AA * HW)     // 1,730,560  (B,A,H,W) planes
#define SPLIT 4               // plane slices per (b,a)
#define THREADS 256           // 8 wave32 waves

// ---------------------------------------------------------------------------
// CDNA5 async global->LDS copy (ASYNCcnt path).  The gfx1250 builtin is
// declared with int* params (AS1 / AS3, printed by clang as __device__ /
// __shared__); clang implicitly addrspace-casts generic pointers for builtin
// args in HIP, so plain (int*) casts are the correct spelling.
// ---------------------------------------------------------------------------
__device__ __forceinline__ void async_copy_b32(const float* g, float* lds_generic) {
#if __has_builtin(__builtin_amdgcn_global_load_async_to_lds_b32)
  __builtin_amdgcn_global_load_async_to_lds_b32((int*)g, (int*)lds_generic, 0, 0);
#else
  unsigned lds_off = (unsigned)(uintptr_t)lds_generic; // low 32 bits = LDS offset
  asm volatile("global_load_async_to_lds_b32 %0, %1, off"
               :: "v"(lds_off), "v"(g) : "memory");
#endif
}

__device__ __forceinline__ void wait_async0() {
#if __has_builtin(__builtin_amdgcn_s_wait_asynccnt)
  __builtin_amdgcn_s_wait_asynccnt(0);
#else
  asm volatile("s_wait_asynccnt 0" ::: "memory");
#endif
}

// ---------------------------------------------------------------------------
// Kernel 1: fused streaming pass.
//   reads  xy(2 planes), wh(2 planes), obj(1 plane)  per (b,a)
//   writes t_xy(copy), t_wh(copy), t_weight(0), t_o_obj(copy),
//          t_o_noobj(IoU-masked obj), t_label(-1)
// IoU>0.6 test is division-free:  inter/max(u,eps) > 0.6  <=>
//   inter > 0.375*(parea + tarea)   (u = parea+tarea-inter, parea>0 always)
// ---------------------------------------------------------------------------
__global__ __launch_bounds__(THREADS)
void region_bulk(const float* __restrict__ xy, const float* __restrict__ wh,
                 const float* __restrict__ obj, const float* __restrict__ truth,
                 const float* __restrict__ biases, float* __restrict__ out) {
  __shared__ float s_tr[TT * 5];
  __shared__ float s_xl[TT], s_xr[TT], s_yt[TT], s_yb[TT], s_car[TT];

  const int tid = threadIdx.x;
  const int bid = blockIdx.x;
  const int s   = bid % SPLIT;
  const int ba  = bid / SPLIT;
  const int a   = ba % AA;
  const int b   = ba / AA;

  // Stage truth[b] (150 floats) into LDS via the async-tensor path.
  if (tid < TT * 5) async_copy_b32(truth + (b * TT * 5 + tid), &s_tr[tid]);
  wait_async0();
  __syncthreads();

  // Precompute per-truth box edges and 0.375*area (invalid -> degenerate box).
  if (tid < TT) {
    float tx = s_tr[tid * 5 + 0], ty = s_tr[tid * 5 + 1];
    float tw = s_tr[tid * 5 + 2], th = s_tr[tid * 5 + 3];
    bool val = (tw > 0.0f) && (th > 0.0f);
    s_xl[tid]  = val ? tx - 0.5f * tw : 0.0f;
    s_xr[tid]  = val ? tx + 0.5f * tw : 0.0f;
    s_yt[tid]  = val ? ty - 0.5f * th : 0.0f;
    s_yb[tid]  = val ? ty + 0.5f * th : 0.0f;
    s_car[tid] = val ? 0.375f * tw * th : 0.0f;
  }
  __syncthreads();

  const float invW = 1.0f / (float)WW;
  const float invH = 1.0f / (float)HH;
  const float sbw  = biases[2 * a]     * invW; // bw[a]/W
  const float sbh  = biases[2 * a + 1] * invH; // bh[a]/H

  const int p0 = (b * 2 * AA + 2 * a) * HW; // xy/wh channel-0 plane base
  const int p1 = p0 + HW;                   // channel-1 plane base
  const int q  = (b * AA + a) * HW;         // obj plane base

  const float4* xy4 = (const float4*)xy;
  const float4* wh4 = (const float4*)wh;
  const float4* ob4 = (const float4*)obj;

  float4* o_txy = (float4*)(out);
  float4* o_twh = (float4*)(out + (size_t)N2);
  float4* o_twt = (float4*)(out + (size_t)2 * N2);
  float4* o_obj = (float4*)(out + (size_t)3 * N2);
  float4* o_no  = (float4*)(out + (size_t)3 * N2 + N1);
  float4* o_lab = (float4*)(out + (size_t)3 * N2 + 2 * (size_t)N1);

  const int NVEC = HW / 4;       // 2704 float4 groups per plane
  const int CH   = NVEC / SPLIT; // 676 per slice
  const int vend = (s + 1) * CH;
  const int p0v = p0 >> 2, p1v = p1 >> 2, qv = q >> 2;

  const float4 zero4 = make_float4(0.0f, 0.0f, 0.0f, 0.0f);
  const float4 neg14 = make_float4(-1.0f, -1.0f, -1.0f, -1.0f);

  for (int v = s * CH + tid; v < vend; v += THREADS) {
    if (v + THREADS < vend) { // global_prefetch_b8 for next trip
      __builtin_prefetch(&xy4[p0v + v + THREADS], 0, 1);
      __builtin_prefetch(&wh4[p0v + v + THREADS], 0, 1);
    }
    float4 x0 = xy4[p0v + v];
    float4 x1 = xy4[p1v + v];
    float4 w0 = wh4[p0v + v];
    float4 w1 = wh4[p1v + v];
    float4 ob = ob4[qv + v];

    const int   cell0 = v * 4;           // W%4==0: never crosses a row
    const float fi0   = (float)(cell0 % WW);
    const float fj    = (float)(cell0 / WW);

    float ax0[4] = {x0.x, x0.y, x0.z, x0.w};
    float ax1[4] = {x1.x, x1.y, x1.z, x1.w};
    float aw0[4] = {w0.x, w0.y, w0.z, w0.w};
    float aw1[4] = {w1.x, w1.y, w1.z, w1.w};
    float aob[4] = {ob.x, ob.y, ob.z, ob.w};

    float pxl[4], pxr[4], pyt[4], pyb[4], thr0[4], m[4];
#pragma unroll
    for (int k = 0; k < 4; ++k) {
      float px = (ax0[k] + (fi0 + (float)k)) * invW;
      float py = (ax1[k] + fj) * invH;
      float pw = expf(aw0[k]) * sbw;
      float ph = expf(aw1[k]) * sbh;
      pxl[k] = px - 0.5f * pw;
      pxr[k] = px + 0.5f * pw;
      pyt[k] = py - 0.5f * ph;
      pyb[k] = py + 0.5f * ph;
      thr0[k] = 0.375f * (pw * ph);
      m[k] = -1.0f;
    }

    for (int t = 0; t < TT; ++t) {
      float xl = s_xl[t], xr = s_xr[t], yt = s_yt[t], yb = s_yb[t], car = s_car[t];
#pragma unroll
      for (int k = 0; k < 4; ++k) {
        float l  = fmaxf(pxl[k], xl);
        float r  = fminf(pxr[k], xr);
        float u  = fmaxf(pyt[k], yt);
        float d  = fminf(pyb[k], yb);
        float iw = fmaxf(r - l, 0.0f);
        float ih = fmaxf(d - u, 0.0f);
        float inter = iw * ih;
        m[k] = fmaxf(m[k], inter - (thr0[k] + car)); // >0  <=>  iou>0.6
      }
    }

    float4 no;
    no.x = (m[0] > 0.0f) ? aob[0] : 0.0f;
    no.y = (m[1] > 0.0f) ? aob[1] : 0.0f;
    no.z = (m[2] > 0.0f) ? aob[2] : 0.0f;
    no.w = (m[3] > 0.0f) ? aob[3] : 0.0f;

    o_txy[p0v + v] = x0;   o_txy[p1v + v] = x1;   // t_xy = xy   (WARMUP=False)
    o_twh[p0v + v] = w0;   o_twh[p1v + v] = w1;   // t_wh = wh
    o_twt[p0v + v] = zero4; o_twt[p1v + v] = zero4;
    o_obj[qv + v] = ob;
    o_no [qv + v] = no;
    o_lab[qv + v] = neg14;
  }
}

// ---------------------------------------------------------------------------
// Kernel 2: per-truth scatter.  One thread per batch, truths applied in index
// order (sequential last-write-wins on duplicate target cells). 960 items.
// ---------------------------------------------------------------------------
__global__ void region_scatter(const float* __restrict__ xy, const float* __restrict__ wh,
                               const float* __restrict__ truth,
                               const float* __restrict__ biases,
                               float* __restrict__ out) {
  const int b = threadIdx.x;
  if (b >= BB) return;

  float bw[AA], bh[AA];
#pragma unroll
  for (int n = 0; n < AA; ++n) { bw[n] = biases[2 * n]; bh[n] = biases[2 * n + 1]; }

  float* o_txy = out;
  float* o_twh = out + (size_t)N2;
  float* o_twt = out + (size_t)2 * N2;
  float* o_obj = out + (size_t)3 * N2;
  float* o_no  = out + (size_t)3 * N2 + N1;
  float* o_lab = out + (size_t)3 * N2 + 2 * (size_t)N1;

  const float EPSF = 1e-12f;

  for (int t = 0; t < TT; ++t) {
    const float* tr = truth + ((size_t)b * TT + t) * 5;
    float tx = tr[0], ty = tr[1], tw = tr[2], th = tr[3], tc = tr[4];
    if (!((tw > 0.0f) && (th > 0.0f))) continue; // gj_w=H -> dropped in reference

    int gi = (int)(tx * (float)WW); gi = min(max(gi, 0), WW - 1);
    int gj = (int)(ty * (float)HH); gj = min(max(gj, 0), HH - 1);

    // argmax anchor IoU (first max wins, matching jnp.argmax)
    float tarea = tw * th;
    int   besta = 0;
    float bestiou = -1.0f;
#pragma unroll
    for (int n = 0; n < AA; ++n) {
      float aw = bw[n] * (1.0f / (float)WW);
      float ah = bh[n] * (1.0f / (float)HH);
      float inter = fminf(tw, aw) * fminf(th, ah);
      float iou = inter / fmaxf(tarea + aw * ah - inter, EPSF);
      if (iou > bestiou) { bestiou = iou; besta = n; }
    }
    float bwa = bw[besta], bha = bh[besta];

    float vxx = tx * (float)WW - (float)gi;
    float vxy = ty * (float)HH - (float)gj;
    float vwx = logf(fmaxf(tw * (float)WW / bwa, EPSF));
    float vwy = logf(fmaxf(th * (float)HH / bha, EPSF));
    float wgt = 1.0f * (2.0f - tarea); // COORD_SCALE = 1

    int cell = gj * WW + gi;
    int p0 = (b * 2 * AA + 2 * besta) * HW + cell;
    int p1 = p0 + HW;
    int qq = (b * AA + besta) * HW + cell;

    // predicted box at the matched cell (from ORIGINAL xy/wh inputs)
    float mx = (xy[p0] + (float)gi) * (1.0f / (float)WW);
    float my = (xy[p1] + (float)gj) * (1.0f / (float)HH);
    float mw = expf(wh[p0]) * bwa * (1.0f / (float)WW);
    float mh = expf(wh[p1]) * bha * (1.0f / (float)HH);

    float l = fmaxf(mx - 0.5f * mw, tx - 0.5f * tw);
    float r = fminf(mx + 0.5f * mw, tx + 0.5f * tw);
    float u = fmaxf(my - 0.5f * mh, ty - 0.5f * th);
    float d = fminf(my + 0.5f * mh, ty + 0.5f * th);
    float inter = fmaxf(r - l, 0.0f) * fmaxf(d - u, 0.0f);
    float iou_m = inter / fmaxf(mw * mh + tarea - inter, EPSF); // RESCORE=True

    o_txy[p0] = vxx;  o_txy[p1] = vxy;
    o_twh[p0] = vwx;  o_twh[p1] = vwy;
    o_twt[p0] = wgt;  o_twt[p1] = wgt;
    o_obj[qq] = iou_m;
    o_no [qq] = iou_m;
    o_lab[qq] = tc;
  }
}

// ---------------------------------------------------------------------------
extern "C" void kernel_launch(void* const* d_in, const int* in_sizes, int n_in,
                              void* d_out, int out_size, void* d_ws, size_t ws_size,
                              hipStream_t stream) {
  (void)in_sizes; (void)n_in; (void)out_size; (void)d_ws; (void)ws_size;
  const float* xy     = (const float*)d_in[0];
  const float* wh     = (const float*)d_in[1];
  const float* obj    = (const float*)d_in[2];
  const float* truth  = (const float*)d_in[3];
  const float* biases = (const float*)d_in[4];
  float* out = (float*)d_out;

  region_bulk<<<dim3(BB * AA * SPLIT), dim3(THREADS), 0, stream>>>(
      xy, wh, obj, truth, biases, out);
  region_scatter<<<dim3(1), dim3(32), 0, stream>>>(xy, wh, truth, biases, out);
}